// MoDRouter_33028298506454
// MI455X (gfx1250) — compile-verified
//
#include <hip/hip_runtime.h>
#include <hip/hip_bf16.h>

#define B_ 4
#define T_ 4096
#define D_ 2048
#define C_ 2048   // capacity = T/2

typedef __attribute__((ext_vector_type(2))) float v2f;
typedef __attribute__((ext_vector_type(8))) float v8f;

// ---------------------------------------------------------------------------
// Kernel 1: router logits via V_WMMA_F32_16X16X4_F32 (full fp32 matrix pipe).
// Each wave computes 16 tokens: A = 16x4 tile of x (lanes 0-15: K0-1,
// lanes 16-31: K2-3 per ISA layout), B = W chunk replicated across all 16
// columns (K rows striped over lane-halves like C/D rows). Accumulate 512
// WMMA steps over K=2048; logits live in column 0 = lanes 0 and 16.
// ---------------------------------------------------------------------------
__global__ __launch_bounds__(256) void mod_score_kernel(
    const float* __restrict__ x, const float* __restrict__ W,
    const float* __restrict__ bias, float* __restrict__ logits_out) {
  __shared__ float w_lds[D_];
  const int tid = threadIdx.x;
#pragma unroll
  for (int i = 0; i < D_ / 256; ++i) w_lds[tid + i * 256] = W[tid + i * 256];
  __syncthreads();

  const int lane = tid & 31;
  const int wave = tid >> 5;                       // 8 waves / block
  const int t_base = blockIdx.x * 128 + wave * 16; // 16 tokens per wave
  const int m = lane & 15;                         // token row within tile
  const int koff = (lane >> 4) * 2;                // 0 (lanes 0-15) or 2

  const float* __restrict__ xrow = x + (size_t)(t_base + m) * D_;
  v8f acc = {};
#pragma unroll 8
  for (int kk = 0; kk < D_; kk += 4) {
    v2f a = *(const v2f*)(xrow + kk + koff);       // A: x[token, kk+koff .. +1]
    v2f bmat = *(const v2f*)(&w_lds[kk + koff]);   // B: W replicated over N
    acc = __builtin_amdgcn_wmma_f32_16x16x4_f32(
        /*neg_a=*/false, a, /*neg_b=*/false, bmat,
        /*c_mod=*/(short)0, acc, /*reuse_a=*/false, /*reuse_b=*/false);
  }
  const float bv = bias[0];
  // Column N=0: lane 0 holds M=0..7 in acc[0..7]; lane 16 holds M=8..15.
  if ((lane & 15) == 0) {
    const int mbase = t_base + (lane >> 4) * 8;
#pragma unroll
    for (int i = 0; i < 8; ++i) logits_out[mbase + i] = acc[i] + bv;
  }
}

// ---------------------------------------------------------------------------
// Kernel 2: exact top-C per batch. MSB-first radix select (4 x 8-bit passes)
// on sign-flipped float keys finds the C-th largest value and how many ties
// to keep; a stable index-order compaction then emits sorted(topk_idx) with
// JAX tie semantics (lowest index first).
// ---------------------------------------------------------------------------
__global__ __launch_bounds__(256) void mod_topk_kernel(
    const float* __restrict__ logits, float* __restrict__ idx_f_out,
    int* __restrict__ idx_i_out) {
  __shared__ unsigned keys[T_];
  __shared__ unsigned hist[256];
  __shared__ unsigned scan_buf[256];
  __shared__ unsigned sh_prefix, sh_rank;

  const int b = blockIdx.x;
  const int tid = threadIdx.x;
  const int PER = T_ / 256;  // 16 contiguous tokens per thread

  for (int i = 0; i < PER; ++i) {
    int t = tid * PER + i;
    unsigned u = __float_as_uint(logits[b * T_ + t]);
    keys[t] = (u & 0x80000000u) ? ~u : (u | 0x80000000u);  // order-preserving
  }
  if (tid == 0) { sh_prefix = 0u; sh_rank = C_; }
  __syncthreads();

  for (int pass = 0; pass < 4; ++pass) {
    const int shift = 24 - pass * 8;
    hist[tid] = 0u;
    __syncthreads();
    const unsigned prefix = sh_prefix;
    const unsigned pmask = (pass == 0) ? 0u : (0xFFFFFFFFu << (32 - pass * 8));
    for (int i = 0; i < PER; ++i) {
      unsigned k = keys[tid * PER + i];
      if ((k & pmask) == prefix) atomicAdd(&hist[(k >> shift) & 0xFFu], 1u);
    }
    __syncthreads();
    if (tid == 0) {
      unsigned r = sh_rank, cum = 0;
      for (int bin = 255; bin >= 0; --bin) {
        unsigned h = hist[bin];
        if (cum + h >= r) {
          sh_rank = r - cum;
          sh_prefix = prefix | ((unsigned)bin << shift);
          break;
        }
        cum += h;
      }
    }
    __syncthreads();
  }
  const unsigned kstar = sh_prefix;      // C-th largest key (threshold)
  const unsigned needed_eq = sh_rank;    // #ties at kstar to keep

  // stable compaction: scan per-thread tie counts, then selected counts
  unsigned cnt_eq = 0;
  for (int i = 0; i < PER; ++i) {
    unsigned k = keys[tid * PER + i];
    if (k == kstar) cnt_eq++;
  }
  scan_buf[tid] = cnt_eq;
  __syncthreads();
  if (tid == 0) {
    unsigned s = 0;
    for (int i = 0; i < 256; ++i) { unsigned v = scan_buf[i]; scan_buf[i] = s; s += v; }
  }
  __syncthreads();
  const unsigned eq_base = scan_buf[tid];

  unsigned sel = 0, eqr = eq_base;
  for (int i = 0; i < PER; ++i) {
    unsigned k = keys[tid * PER + i];
    if (k > kstar) sel++;
    else if (k == kstar) { if (eqr < needed_eq) sel++; eqr++; }
  }
  __syncthreads();
  scan_buf[tid] = sel;
  __syncthreads();
  if (tid == 0) {
    unsigned s = 0;
    for (int i = 0; i < 256; ++i) { unsigned v = scan_buf[i]; scan_buf[i] = s; s += v; }
  }
  __syncthreads();
  unsigned pos = scan_buf[tid];

  eqr = eq_base;
  for (int i = 0; i < PER; ++i) {
    int t = tid * PER + i;
    unsigned k = keys[t];
    bool take = false;
    if (k > kstar) take = true;
    else if (k == kstar) { take = (eqr < needed_eq); eqr++; }
    if (take) {
      idx_i_out[b * C_ + pos] = t;
      idx_f_out[b * C_ + pos] = (float)t;  // exact: t < 4096
      pos++;
    }
  }
}

// ---------------------------------------------------------------------------
// Kernel 3: gather selected rows, b128 loads/stores. One block per row.
// ---------------------------------------------------------------------------
__global__ __launch_bounds__(256) void mod_gather_kernel(
    const float* __restrict__ x, const int* __restrict__ idx,
    float* __restrict__ out) {
  const int row = blockIdx.x;          // 0 .. B*C-1
  const int b = row >> 11;             // / C_
  const int t = idx[row];
  const float4* __restrict__ src =
      (const float4*)(x + ((size_t)b * T_ + (size_t)t) * D_);
  float4* __restrict__ dst = (float4*)(out + (size_t)row * D_);
  const int tid = threadIdx.x;
#pragma unroll
  for (int i = 0; i < 2; ++i) dst[tid + i * 256] = src[tid + i * 256];
}

extern "C" void kernel_launch(void* const* d_in, const int* in_sizes, int n_in,
                              void* d_out, int out_size, void* d_ws, size_t ws_size,
                              hipStream_t stream) {
  const float* x = (const float*)d_in[0];     // (B,T,D) fp32
  const float* W = (const float*)d_in[1];     // (1,D)  fp32
  const float* bias = (const float*)d_in[2];  // (1,)   fp32

  float* out = (float*)d_out;
  float* out_sel    = out;                               // B*C*D
  float* out_idx    = out + (size_t)B_ * C_ * D_;        // B*C (as float values)
  float* out_logits = out_idx + (size_t)B_ * C_;         // B*T
  int* ws_idx = (int*)d_ws;                              // B*C ints

  mod_score_kernel<<<(B_ * T_) / 128, 256, 0, stream>>>(x, W, bias, out_logits);
  mod_topk_kernel<<<B_, 256, 0, stream>>>(out_logits, out_idx, ws_idx);
  mod_gather_kernel<<<B_ * C_, 256, 0, stream>>>(x, ws_idx, out_sel);
}